// UnitaryPMMSequence_61924838474105
// MI455X (gfx1250) — compile-verified
//
#include <hip/hip_runtime.h>

// MI455X / gfx1250 fused pipeline:
//   GEMM1 (WMMA f32 16x16x4): Htile = x_tile @ Wherm + H0herm
//   Batched 16x16 complex Hermitian Jacobi eigensolve in LDS (wave32, 16 lanes/batch)
//   Projector G = v v^H onto 4 lowest eigenvectors
//   GEMM2 (WMMA f32 16x16x4): Y = Gflat @ Sv^T + bias
// HBM traffic ~32MB total (x in, y out); WhT/Sv (256KB) stay in 192MB L2.
// f32 WMMA (not fp16/fp8): GEMM1 output feeds an eigensolver, precision must hold.
// Jacobi phase: per-batch work confined to 16 lanes of one wave32 -> wave-level
// scheduling barriers only (DS ops from one wave are in-order); all complex LDS
// traffic is 64-bit and bank-conflict-free via XOR swizzle.

typedef __attribute__((ext_vector_type(2))) float v2f;
typedef __attribute__((ext_vector_type(8))) float v8f;

#define NSWEEP 8

__device__ __forceinline__ v2f cmul(v2f a, v2f b) {   // a*b
  v2f r; r.x = a.x * b.x - a.y * b.y; r.y = a.x * b.y + a.y * b.x; return r;
}
__device__ __forceinline__ v2f cmulc(v2f a, v2f b) {  // conj(a)*b
  v2f r; r.x = a.x * b.x + a.y * b.y; r.y = a.x * b.y - a.y * b.x; return r;
}

// ws float layout: WhT[512][64] @0, Sv[64][512] @32768, H0f[512] @65536
__global__ void upmm_prep_kernel(const float* __restrict__ H0,
                                 const float* __restrict__ Hin,
                                 const float* __restrict__ S,
                                 float* __restrict__ ws) {
  int idx = blockIdx.x * blockDim.x + threadIdx.x;
  if (idx < 32768) {                      // WhT[j][l] = herm(H_inputs) flattened, transposed
    int l = idx >> 9, j = idx & 511;
    int p = j & 1, m = (j >> 1) & 15, n = j >> 5;
    const float* a = Hin + (l * 256 + n * 16 + m) * 2;
    const float* b = Hin + (l * 256 + m * 16 + n) * 2;
    ws[j * 64 + l] = (p == 0) ? 0.5f * (a[0] + b[0]) : 0.5f * (a[1] - b[1]);
  } else if (idx < 65536) {               // Sv[k][j] = herm(S) flattened
    int i2 = idx - 32768;
    int k = i2 >> 9, j = i2 & 511;
    int p = j & 1, m = (j >> 1) & 15, n = j >> 5;
    const float* a = S + (k * 256 + n * 16 + m) * 2;
    const float* b = S + (k * 256 + m * 16 + n) * 2;
    ws[idx] = (p == 0) ? 0.5f * (a[0] + b[0]) : 0.5f * (a[1] - b[1]);
  } else if (idx < 66048) {               // H0f = herm(H0) flattened
    int j = idx - 65536;
    int p = j & 1, m = (j >> 1) & 15, n = j >> 5;
    const float* a = H0 + (n * 16 + m) * 2;
    const float* b = H0 + (m * 16 + n) * 2;
    ws[idx] = (p == 0) ? 0.5f * (a[0] + b[0]) : 0.5f * (a[1] - b[1]);
  }
}

// XOR-swizzled complex-entry index (units of v2f): bank-conflict-free for both
// row-parallel (lane=n) and column-parallel (lane=m) Jacobi phases, 64b accesses.
__device__ __forceinline__ int cidx(int n, int m) { return n * 16 + (m ^ n); }

__global__ __launch_bounds__(256)
void upmm_main_kernel(const float* __restrict__ x,
                      const float* __restrict__ ws,
                      const float* __restrict__ bias,
                      float* __restrict__ out) {
  __shared__ v2f sH[16 * 256];  // 16 batches x 16x16 complex (swizzled), later G
  __shared__ v2f sV[16 * 256];  // eigenvectors; later GEMM2 partial buffer

  const float* WhT = ws;
  const float* Sv  = ws + 32768;
  const float* H0f = ws + 65536;

  const int tid    = threadIdx.x;
  const int wave   = tid >> 5;
  const int lane   = tid & 31;
  const int laneLo = lane & 15;
  const int laneHi = lane >> 4;
  const int bbase  = blockIdx.x * 16;

  // ---------------- Phase A: Htile = x @ Wh + H0f  (WMMA f32 16x16x4) -------
  {
    v8f acc[4] = {};
    const float* xrow = x + (bbase + laneLo) * 64;
    for (int kb = 0; kb < 16; ++kb) {
      const int k0 = kb * 4 + laneHi * 2;   // A/B element k = k0 + j (j=0,1)
      const v2f a = *(const v2f*)(xrow + k0);
#pragma unroll
      for (int i = 0; i < 4; ++i) {
        const int col = (wave * 4 + i) * 16 + laneLo;
        const v2f b = *(const v2f*)(WhT + col * 64 + k0);
        acc[i] = __builtin_amdgcn_wmma_f32_16x16x4_f32(
            false, a, false, b, (short)0, acc[i], false, false);
      }
    }
    float* sHf = (float*)sH;
#pragma unroll
    for (int i = 0; i < 4; ++i) {
      const int col = (wave * 4 + i) * 16 + laneLo;   // flat (n,m,part) index
      const int part = col & 1, ci = col >> 1;
      const int m2 = ci & 15, n2 = ci >> 4;
      const float h0v = H0f[col];
#pragma unroll
      for (int j = 0; j < 8; ++j) {
        const int bm = j + laneHi * 8;                // batch row in tile
        sHf[bm * 512 + cidx(n2, m2) * 2 + part] = acc[i][j] + h0v;
      }
    }
  }
  __syncthreads();

  // ---------------- Phase B: Jacobi eigensolve, 16 lanes per batch ----------
  const int b = wave * 2 + laneHi;   // batch in tile served by this half-wave
  const int r = laneLo;              // row owned by this lane
  v2f* Hb = sH + b * 256;
  v2f* Vb = sV + b * 256;

#pragma unroll
  for (int m = 0; m < 16; ++m) {     // V = I
    v2f e; e.x = (m == r) ? 1.0f : 0.0f; e.y = 0.0f;
    Vb[cidx(r, m)] = e;
  }
  __builtin_amdgcn_wave_barrier();

#pragma unroll 1
  for (int sweep = 0; sweep < NSWEEP; ++sweep) {
#pragma unroll 1
    for (int p = 0; p < 15; ++p) {
#pragma unroll 1
      for (int q = p + 1; q < 16; ++q) {
        // rotation params (broadcast LDS reads; wave DS ops are in-order)
        const float app = Hb[cidx(p, p)].x;
        const float aqq = Hb[cidx(q, q)].x;
        const v2f apq = Hb[cidx(p, q)];
        const float mag2 = apq.x * apq.x + apq.y * apq.y;
        float c = 1.0f, s = 0.0f;
        v2f u; u.x = 1.0f; u.y = 0.0f;
        if (mag2 > 1e-24f) {
          const float mag = sqrtf(mag2);
          u.x = apq.x / mag; u.y = -apq.y / mag;          // u = conj(apq)/|apq|
          const float tau = (aqq - app) / (2.0f * mag);
          const float t = copysignf(1.0f, tau) / (fabsf(tau) + sqrtf(1.0f + tau * tau));
          c = 1.0f / sqrtf(1.0f + t * t);
          s = t * c;
        }
        v2f su; su.x = s * u.x; su.y = s * u.y;
        v2f cu; cu.x = c * u.x; cu.y = c * u.y;
        __builtin_amdgcn_wave_barrier();
        // column update: lane r updates H(r,p),H(r,q) and V(r,p),V(r,q)
        {
          const int ip = cidx(r, p), iq = cidx(r, q);
          const v2f hp = Hb[ip], hq = Hb[iq];
          const v2f tp = cmul(su, hq), tq = cmul(cu, hq);
          v2f np, nq;
          np.x = c * hp.x - tp.x; np.y = c * hp.y - tp.y;
          nq.x = s * hp.x + tq.x; nq.y = s * hp.y + tq.y;
          Hb[ip] = np; Hb[iq] = nq;
          const v2f vp = Vb[ip], vq = Vb[iq];
          const v2f wp = cmul(su, vq), wq = cmul(cu, vq);
          v2f mp, mq;
          mp.x = c * vp.x - wp.x; mp.y = c * vp.y - wp.y;
          mq.x = s * vp.x + wq.x; mq.y = s * vp.y + wq.y;
          Vb[ip] = mp; Vb[iq] = mq;
        }
        __builtin_amdgcn_wave_barrier();
        // row update: lane r updates H(p,r),H(q,r)  (conj(u) enters)
        {
          const int ip = cidx(p, r), iq = cidx(q, r);
          const v2f hp = Hb[ip], hq = Hb[iq];
          const v2f tp = cmulc(su, hq), tq = cmulc(cu, hq);
          v2f np, nq;
          np.x = c * hp.x - tp.x; np.y = c * hp.y - tp.y;
          nq.x = s * hp.x + tq.x; nq.y = s * hp.y + tq.y;
          Hb[ip] = np; Hb[iq] = nq;
        }
        __builtin_amdgcn_wave_barrier();
      }
    }
  }

  // ---- select 4 smallest eigenvalues, G = v v^H (overwrites H in place) ----
  unsigned mask = 0;
  {
    float d[16];
#pragma unroll
    for (int j2 = 0; j2 < 16; ++j2) d[j2] = Hb[cidx(j2, j2)].x;
    for (int cc = 0; cc < 16; ++cc) {
      int rank = 0;
      for (int j2 = 0; j2 < 16; ++j2)
        rank += (d[j2] < d[cc]) || (d[j2] == d[cc] && j2 < cc);
      if (rank < 4) mask |= (1u << cc);
    }
  }
  __builtin_amdgcn_wave_barrier();
  for (int m = 0; m < 16; ++m) {
    v2f g; g.x = 0.0f; g.y = 0.0f;
    for (int cc = 0; cc < 16; ++cc) {
      if (mask & (1u << cc)) {
        const v2f vr = Vb[cidx(r, cc)];
        const v2f vm = Vb[cidx(m, cc)];
        const v2f t = cmulc(vm, vr);        // v[r]*conj(v[m])
        g.x += t.x; g.y += t.y;
      }
    }
    Hb[cidx(r, m)] = g;
  }
  __syncthreads();   // phase C reads all 16 batches across waves

  // ---------------- Phase C: Y = Gflat @ Sv^T + bias  (WMMA f32 16x16x4) ----
  {
    const int nt = wave & 3;        // N-tile of Q=64 (4 tiles of 16)
    const int khalf = wave >> 2;    // K split in two halves of 256
    const int col = nt * 16 + laneLo;
    const float* svrow = Sv + col * 512;
    v8f acc = {};
    for (int kb = 0; kb < 64; ++kb) {
      const int kbase = khalf * 256 + kb * 4 + laneHi * 2;   // even
      const int ci = kbase >> 1;
      const int m2 = ci & 15, n2 = ci >> 4;
      const v2f a = sH[laneLo * 256 + cidx(n2, m2)];  // Gflat[batch=laneLo][kbase..+1]
      const v2f bb = *(const v2f*)(svrow + kbase);
      acc = __builtin_amdgcn_wmma_f32_16x16x4_f32(
          false, a, false, bb, (short)0, acc, false, false);
    }
    float* partial = (float*)sV;    // reuse: 4 tiles * 256 floats
    if (khalf == 1) {
#pragma unroll
      for (int j = 0; j < 8; ++j) partial[nt * 256 + j * 32 + lane] = acc[j];
    }
    __syncthreads();
    if (khalf == 0) {
      const float bv = bias[col];
#pragma unroll
      for (int j = 0; j < 8; ++j) {
        const int bm = j + laneHi * 8;
        out[(bbase + bm) * 64 + col] =
            acc[j] + partial[nt * 256 + j * 32 + lane] + bv;
      }
    }
  }
}

extern "C" void kernel_launch(void* const* d_in, const int* in_sizes, int n_in,
                              void* d_out, int out_size, void* d_ws, size_t ws_size,
                              hipStream_t stream) {
  (void)in_sizes; (void)n_in; (void)out_size; (void)ws_size;
  const float* x    = (const float*)d_in[0];
  const float* H0   = (const float*)d_in[1];   // complex64 -> float pairs
  const float* Hin  = (const float*)d_in[2];
  const float* S    = (const float*)d_in[3];
  const float* bias = (const float*)d_in[4];
  float* ws  = (float*)d_ws;
  float* out = (float*)d_out;

  upmm_prep_kernel<<<(66048 + 255) / 256, 256, 0, stream>>>(H0, Hin, S, ws);
  upmm_main_kernel<<<65536 / 16, 256, 0, stream>>>(x, ws, bias, out);
}